// SpanEmbedder_46385646797402
// MI455X (gfx1250) — compile-verified
//
#include <hip/hip_runtime.h>
#include <hip/hip_bf16.h>

// Problem constants (from reference)
#define T_TOK   65536
#define D_DIM   768
#define S_SEG   8192
#define FLAT_N  65536

typedef float v2f __attribute__((ext_vector_type(2)));
typedef float v8f __attribute__((ext_vector_type(8)));

// ---------------------------------------------------------------------------
// Generic f32 WMMA GEMM: C[M,N] = A[M,K] @ B[K,N] (+ bias[N])
// Row-major everything. Each wave computes a 32(M) x 64(N) tile.
// K stepped by 4 using V_WMMA_F32_16X16X4_F32.
// M % 32 == 0, N % 64 == 0, K % 4 == 0 (true for all call sites here).
// ---------------------------------------------------------------------------
__global__ __launch_bounds__(256) void gemm_f32_wmma(
    const float* __restrict__ A, const float* __restrict__ B,
    const float* __restrict__ bias, float* __restrict__ C,
    int M, int N, int K, int useBias)
{
    const int lane      = threadIdx.x & 31;
    const int waveInBlk = threadIdx.x >> 5;
    const int wave      = blockIdx.x * 8 + waveInBlk;

    const int tilesN = N >> 6;                 // 64-wide N tiles
    const int tileM  = (wave / tilesN) << 5;   // 32-high M tiles
    const int tileN  = (wave % tilesN) << 6;
    if (tileM >= M) return;                    // wave-uniform exit

    const int laneHalf = lane >> 4;            // 0..1 (selects K pair / M+8)
    const int lane16   = lane & 15;

    v8f acc[2][4] = {};                        // zero-init accumulators

    // A fragment addressing (ISA 7.12.2, 32-bit A 16x4):
    //   lane l (l<16): v0=A[m][0], v1=A[m][1];  lane l+16: v0=A[m][2], v1=A[m][3]
    const float* Ap0 = A + (size_t)(tileM + lane16) * K + 2 * laneHalf;
    const float* Ap1 = Ap0 + (size_t)16 * K;

    // B fragment addressing (4x16): v0 lane n = B[0][n] (half0) / B[2][n] (half1),
    //                               v1 lane n = B[1][n] / B[3][n]
    // Per-lane pointer; everything else becomes immediate offsets.
    const float* Bp = B + (size_t)(2 * laneHalf) * N + tileN + lane16;
    const size_t bStep = (size_t)4 * N;        // floats advanced per K-step

    #pragma unroll 2
    for (int k = 0; k < K; k += 4) {
        v2f a0, a1;
        a0.x = Ap0[0];  a0.y = Ap0[1];
        a1.x = Ap1[0];  a1.y = Ap1[1];
        Ap0 += 4;  Ap1 += 4;

        #pragma unroll
        for (int j = 0; j < 4; ++j) {
            v2f b;
            b.x = Bp[16 * j];          // row k'(=0/2), imm offset 0/64/128/192 B
            b.y = Bp[16 * j + N];      // row k'+1,     imm offset +3072 B
            acc[0][j] = __builtin_amdgcn_wmma_f32_16x16x4_f32(
                false, a0, false, b, (short)0, acc[0][j], false, false);
            acc[1][j] = __builtin_amdgcn_wmma_f32_16x16x4_f32(
                false, a1, false, b, (short)0, acc[1][j], false, false);
        }
        Bp += bStep;
    }

    // C/D layout: vgpr i, lanes 0-15 -> (M = i, N = lane); lanes 16-31 -> M = i+8
    #pragma unroll
    for (int ti = 0; ti < 2; ++ti) {
        const int rowBase = tileM + 16 * ti + 8 * laneHalf;
        #pragma unroll
        for (int j = 0; j < 4; ++j) {
            const int col = tileN + 16 * j + lane16;
            const float bv = useBias ? bias[col] : 0.0f;
            #pragma unroll
            for (int i = 0; i < 8; ++i) {
                C[(size_t)(rowBase + i) * N + col] = acc[ti][j][i] + bv;
            }
        }
    }
}

// ---------------------------------------------------------------------------
// b_comb[n] = sum_k b_tok[k] * W_span[k][n] + b_span[n]
// ---------------------------------------------------------------------------
__global__ __launch_bounds__(256) void bias_combine_kernel(
    const float* __restrict__ b_tok, const float* __restrict__ W_span,
    const float* __restrict__ b_span, float* __restrict__ b_comb)
{
    const int n = blockIdx.x * blockDim.x + threadIdx.x;
    if (n >= D_DIM) return;
    float acc = b_span[n];
    for (int k = 0; k < D_DIM; ++k)
        acc += b_tok[k] * W_span[(size_t)k * D_DIM + n];
    b_comb[n] = acc;
}

// ---------------------------------------------------------------------------
// Segment mean of raw token embeddings.
// segment_ids is sorted; block b handles span b: binary-search [begin,end),
// each of 256 threads accumulates 3 consecutive columns (one b96 load per
// row; block-level access is the full contiguous 3 KB row).
// ---------------------------------------------------------------------------
__global__ __launch_bounds__(256) void seg_mean_kernel(
    const float* __restrict__ emb, const int* __restrict__ span_idx,
    const int* __restrict__ seg_ids, float* __restrict__ pooled)
{
    const int seg = blockIdx.x;

    // lower_bound(seg)
    int lo = 0, hi = FLAT_N;
    while (lo < hi) { int mid = (lo + hi) >> 1; if (seg_ids[mid] < seg) lo = mid + 1; else hi = mid; }
    const int begin = lo;
    // lower_bound(seg+1)
    hi = FLAT_N;
    while (lo < hi) { int mid = (lo + hi) >> 1; if (seg_ids[mid] < seg + 1) lo = mid + 1; else hi = mid; }
    const int end = lo;

    const int c = 3 * threadIdx.x;   // 768 = 3 * 256
    float a0 = 0.0f, a1 = 0.0f, a2 = 0.0f;
    for (int j = begin; j < end; ++j) {
        const float* r = emb + (size_t)span_idx[j] * D_DIM + c;
        a0 += r[0];
        a1 += r[1];
        a2 += r[2];
    }
    const float inv = 1.0f / fmaxf((float)(end - begin), 1.0f);
    float* o = pooled + (size_t)seg * D_DIM + c;
    o[0] = a0 * inv;
    o[1] = a1 * inv;
    o[2] = a2 * inv;
}

// ---------------------------------------------------------------------------
extern "C" void kernel_launch(void* const* d_in, const int* in_sizes, int n_in,
                              void* d_out, int out_size, void* d_ws, size_t ws_size,
                              hipStream_t stream)
{
    const float* emb    = (const float*)d_in[0];  // [T, D]
    const float* W_tok  = (const float*)d_in[1];  // [D, D]
    const float* b_tok  = (const float*)d_in[2];  // [D]
    const float* W_span = (const float*)d_in[3];  // [D, D]
    const float* b_span = (const float*)d_in[4];  // [D]
    const int*   s_idx  = (const int*)d_in[5];    // [FLAT]
    const int*   seg_id = (const int*)d_in[6];    // [FLAT] (sorted)
    float* out = (float*)d_out;                   // [S, D]

    // Workspace layout (floats): W_comb[D*D] | b_comb[1024 pad] | pooled[S*D]
    float* W_comb = (float*)d_ws;
    float* b_comb = W_comb + (size_t)D_DIM * D_DIM;
    float* pooled = b_comb + 1024;

    // 1) W_comb = W_tok @ W_span   (768x768x768)
    {
        const int waves  = (D_DIM / 32) * (D_DIM / 64);
        const int blocks = (waves + 7) / 8;
        gemm_f32_wmma<<<blocks, 256, 0, stream>>>(
            W_tok, W_span, nullptr, W_comb, D_DIM, D_DIM, D_DIM, 0);
    }
    // 2) b_comb = b_tok @ W_span + b_span
    bias_combine_kernel<<<(D_DIM + 255) / 256, 256, 0, stream>>>(
        b_tok, W_span, b_span, b_comb);
    // 3) pooled = segment_mean(emb[span_idx])   [S, D]
    seg_mean_kernel<<<S_SEG, 256, 0, stream>>>(emb, s_idx, seg_id, pooled);
    // 4) out = pooled @ W_comb + b_comb   (8192x768x768)
    {
        const int waves  = (S_SEG / 32) * (D_DIM / 64);
        const int blocks = (waves + 7) / 8;
        gemm_f32_wmma<<<blocks, 256, 0, stream>>>(
            pooled, W_comb, b_comb, out, S_SEG, D_DIM, D_DIM, 1);
    }
}